// ImplicitSurface_49100066127891
// MI455X (gfx1250) — compile-verified
//
#include <hip/hip_runtime.h>
#include <math.h>

#define S      128     // samples per ray
#define SI     127     // intervals per ray
#define NI     64      // n_importance
#define RPB    16      // rays per block
#define TPB    256     // 8 wave32
#define LSTR   132     // padded LDS row stride (floats) -> conflict-free, 16B aligned rows

typedef __attribute__((ext_vector_type(2))) float v2f;
typedef __attribute__((ext_vector_type(8))) float v8f;
typedef __attribute__((ext_vector_type(4))) int   v4i;

#if defined(__gfx1250__) && __has_builtin(__builtin_amdgcn_global_load_async_to_lds_b128)
#define HAVE_ASYNC_LDS 1
typedef __attribute__((address_space(1))) v4i* as1_v4i_p;   // global (AS1) v4i*
typedef __attribute__((address_space(3))) v4i* as3_v4i_p;   // LDS    (AS3) v4i*
#else
#define HAVE_ASYNC_LDS 0
#endif

__device__ __forceinline__ float sigmoidf(float x) {
  return 1.0f / (1.0f + __expf(-x));
}

__global__ __launch_bounds__(TPB) void neus_upsample_kernel(
    const float* __restrict__ rays_o, const float* __restrict__ rays_d,
    const float* __restrict__ z_vals, const float* __restrict__ sdf_g,
    const float* __restrict__ vm_g,   const float* __restrict__ inv_s_p,
    float* __restrict__ out)
{
  __shared__ __align__(16) float s_z[RPB * LSTR];   // bins (kept for stage C)
  __shared__ __align__(16) float s_f[RPB * LSTR];   // sdf
  __shared__ __align__(16) float s_m[RPB * LSTR];   // voxel mask
  __shared__ __align__(16) float s_w[RPB * LSTR];   // pdf, then overwritten with cdf
  __shared__ float s_bs[RPB * 8];                   // per-16-block sums
  __shared__ float s_cy[RPB * 8];                   // exclusive carries

  const int t    = threadIdx.x;
  const int wave = t >> 5;
  const int lane = t & 31;
  const int ray0 = blockIdx.x * RPB;
  const float inv_s = inv_s_p[0];

  // ---------------- stage 0: HBM -> LDS staging ----------------
  // gfx1250 path: direct async memory->LDS (no VGPR round trip, ASYNCcnt).
  {
    const float* gz = z_vals + (size_t)ray0 * S;
    const float* gf = sdf_g  + (size_t)ray0 * S;
    const float* gm = vm_g   + (size_t)ray0 * S;
#if HAVE_ASYNC_LDS
#pragma unroll
    for (int it = 0; it < 2; ++it) {
      int e4   = t + it * TPB;          // 0..511 float4 slots (16 rays * 32)
      int r    = e4 >> 5;
      int p    = (e4 & 31) << 2;
      int goff = e4 << 2;               // float offset into the block's slab
      __builtin_amdgcn_global_load_async_to_lds_b128(
          (as1_v4i_p)(gz + goff), (as3_v4i_p)&s_z[r * LSTR + p], 0, 0);
      __builtin_amdgcn_global_load_async_to_lds_b128(
          (as1_v4i_p)(gf + goff), (as3_v4i_p)&s_f[r * LSTR + p], 0, 0);
      __builtin_amdgcn_global_load_async_to_lds_b128(
          (as1_v4i_p)(gm + goff), (as3_v4i_p)&s_m[r * LSTR + p], 0, 0);
    }
#if __has_builtin(__builtin_amdgcn_s_wait_asynccnt)
    __builtin_amdgcn_s_wait_asynccnt(0);
#else
    asm volatile("s_wait_asynccnt 0x0" ::: "memory");
#endif
#else
    const float4* gz4 = (const float4*)gz;
    const float4* gf4 = (const float4*)gf;
    const float4* gm4 = (const float4*)gm;
#pragma unroll
    for (int it = 0; it < 2; ++it) {
      int e4 = t + it * TPB;
      int r  = e4 >> 5;
      int p  = (e4 & 31) << 2;
      *(float4*)&s_z[r * LSTR + p] = gz4[e4];
      *(float4*)&s_f[r * LSTR + p] = gf4[e4];
      *(float4*)&s_m[r * LSTR + p] = gm4[e4];
    }
#endif
  }
  __syncthreads();

  // ---------------- stage A: weights -> normalized pdf (two rays per wave) -------
#pragma unroll
  for (int rr = 0; rr < 2; ++rr) {
    const int r = wave * 2 + rr;
    const size_t rg = (size_t)(ray0 + r) * 3;
    const float ox = rays_o[rg+0], oy = rays_o[rg+1], oz = rays_o[rg+2];
    const float dx = rays_d[rg+0], dy = rays_d[rg+1], dz = rays_d[rg+2];

    const float* zr = &s_z[r * LSTR];
    const float* fr = &s_f[r * LSTR];
    const float* mr = &s_m[r * LSTR];

    const int p0 = lane << 2;                     // 4 samples / lane
    float4 z4 = *(const float4*)(zr + p0);
    float4 f4 = *(const float4*)(fr + p0);
    float4 m4 = *(const float4*)(mr + p0);
    const bool hasN = (p0 + 4) < S;
    const float zN = hasN ? zr[p0+4] : z4.w;
    const float fN = hasN ? fr[p0+4] : f4.w;
    const float mN = hasN ? mr[p0+4] : m4.w;

    float zz[5] = {z4.x, z4.y, z4.z, z4.w, zN};
    float ff[5] = {f4.x, f4.y, f4.z, f4.w, fN};
    float mm[5] = {m4.x, m4.y, m4.z, m4.w, mN};
    float rad[5];
#pragma unroll
    for (int k = 0; k < 5; ++k) {
      float px = ox + dx*zz[k], py = oy + dy*zz[k], pz = oz + dz*zz[k];
      rad[k] = sqrtf(px*px + py*py + pz*pz);
    }

    float cr[4];
#pragma unroll
    for (int k = 0; k < 4; ++k)
      cr[k] = (ff[k+1] - ff[k]) / (zz[k+1] - zz[k] + 1e-5f);

    // raw-cos of previous interval (reference prev_cos), prev_cos[0] = 0
    float up  = __shfl_up(cr[3], 1, 32);
    float pc0 = (lane == 0) ? 0.0f : up;
    float pc[4] = {pc0, cr[0], cr[1], cr[2]};

    float alpha[4];
#pragma unroll
    for (int k = 0; k < 4; ++k) {
      float c = fminf(pc[k], cr[k]);
      c = fminf(fmaxf(c, -1000.0f), 0.0f);
      float ins = (((rad[k] < 1.0f) || (rad[k+1] < 1.0f)) &&
                   (mm[k] * mm[k+1] > 0.0f)) ? 1.0f : 0.0f;
      c *= ins;
      float mid  = 0.5f * (ff[k] + ff[k+1]);
      float dist = zz[k+1] - zz[k];
      float pcdf = sigmoidf((mid - c * dist * 0.5f) * inv_s);
      float ncdf = sigmoidf((mid + c * dist * 0.5f) * inv_s);
      alpha[k] = (pcdf - ncdf + 1e-5f) / (pcdf + 1e-5f);
      if (p0 + k >= SI) alpha[k] = 0.0f;          // interval 127 doesn't exist
    }

    // exclusive cumprod of (1 - alpha + 1e-7): multiplicative wave scan
    float ml[4];
#pragma unroll
    for (int k = 0; k < 4; ++k)
      ml[k] = (p0 + k < SI) ? (1.0f - alpha[k] + 1e-7f) : 1.0f;
    float e0 = 1.0f, e1 = ml[0], e2 = e1 * ml[1], e3 = e2 * ml[2];
    float sc = e3 * ml[3];
#pragma unroll
    for (int d = 1; d < 32; d <<= 1) {
      float o = __shfl_up(sc, d, 32);
      if (lane >= d) sc *= o;
    }
    float off = __shfl_up(sc, 1, 32);
    if (lane == 0) off = 1.0f;

    float w[4], lsum = 0.0f;
    float tr[4] = {off * e0, off * e1, off * e2, off * e3};
#pragma unroll
    for (int k = 0; k < 4; ++k) {
      w[k] = (p0 + k < SI) ? (alpha[k] * tr[k] + 1e-5f) : 0.0f;  // slot 127 pdf = 0
      lsum += w[k];
    }
#pragma unroll
    for (int d = 16; d >= 1; d >>= 1) lsum += __shfl_xor(lsum, d, 32);
    float inv = 1.0f / lsum;
    *(float4*)&s_w[r * LSTR + p0] =
        make_float4(w[0]*inv, w[1]*inv, w[2]*inv, w[3]*inv);
  }
  __syncthreads();

  // ---------------- stage B: inclusive CDF cumsum via V_WMMA_F32_16X16X4_F32 ----
  // D(16 rays x 16 pos) = sum_j A_j(16x4) * B_j(4x16), B_j[k][n] = (4j+k <= n).
  // A layout (ISA 7.12.2): lane m<16 -> {K0,K1}, lane m+16 -> {K2,K3}.
  // B layout: vgpr0 -> rows {K0 | K2}, vgpr1 -> rows {K1 | K3}, N = lane%16.
  float tile[8];
  {
    const int b    = wave;            // position block 0..7 (16 positions each)
    const int n    = lane & 15;
    const int half = lane >> 4;
    v8f c = {0.f,0.f,0.f,0.f,0.f,0.f,0.f,0.f};
#pragma unroll
    for (int j = 0; j < 4; ++j) {
      const int k0 = 4*j + 2*half;
      v2f a, bb;
      a.x  = s_w[n * LSTR + 16*b + k0];
      a.y  = s_w[n * LSTR + 16*b + k0 + 1];
      bb.x = (k0     <= n) ? 1.0f : 0.0f;
      bb.y = (k0 + 1 <= n) ? 1.0f : 0.0f;
      c = __builtin_amdgcn_wmma_f32_16x16x4_f32(
              false, a, false, bb, (short)0, c, false, false);
    }
    union { v8f v; float f[8]; } cu; cu.v = c;
#pragma unroll
    for (int r8 = 0; r8 < 8; ++r8) tile[r8] = cu.f[r8];

    if (n == 15) {                    // column 15 = block total per ray
#pragma unroll
      for (int r8 = 0; r8 < 8; ++r8)
        s_bs[(r8 + 8*half) * 8 + b] = tile[r8];
    }
  }
  __syncthreads();

  if (t < RPB * 8) {                  // exclusive carry across the 8 blocks
    const int mray = t >> 3, bb = t & 7;
    float acc = 0.0f;
    for (int x = 0; x < bb; ++x) acc += s_bs[mray * 8 + x];
    s_cy[t] = acc;
  }
  __syncthreads();

  {
    const int b    = wave;
    const int n    = lane & 15;
    const int half = lane >> 4;
    const int pos  = 16*b + n;        // inclusive prefix at pos -> cdf[pos+1]
#pragma unroll
    for (int r8 = 0; r8 < 8; ++r8) {
      const int mray = r8 + 8*half;
      if (pos < SI)
        s_w[mray * LSTR + pos + 1] = tile[r8] + s_cy[mray * 8 + b];
    }
  }
  if (t < RPB) s_w[t * LSTR] = 0.0f;  // cdf[0] = 0
  __syncthreads();

  // ---------------- stage C: branchless searchsorted(right) + lerp --------------
#pragma unroll
  for (int q = 0; q < 4; ++q) {
    const int task = t + q * TPB;     // 16 rays * 64 samples
    const int r    = task >> 6;
    const int k    = task & 63;
    const float u  = (k + 0.5f) * (1.0f / 64.0f);
    const float* cdf = &s_w[r * LSTR];
    // largest idx with cdf[idx] <= u (valid: cdf[0] = 0 < u); fixed 7 steps.
    int lo = 0;
#pragma unroll
    for (int stp = 64; stp >= 1; stp >>= 1) {
      int cand = lo + stp;
      if (cand <= S - 1 && cdf[cand] <= u) lo = cand;
    }
    const int below = lo;             // = searchsorted_right - 1
    int above = lo + 1; if (above > S - 1) above = S - 1;
    const float cb = cdf[below], ca = cdf[above];
    const float zb = s_z[r * LSTR + below], za = s_z[r * LSTR + above];
    float denom = ca - cb;
    denom = (denom < 1e-5f) ? 1.0f : denom;
    const float tt = (u - cb) / denom;
    out[(size_t)(ray0 + r) * NI + k] = zb + tt * (za - zb);
  }
}

extern "C" void kernel_launch(void* const* d_in, const int* in_sizes, int n_in,
                              void* d_out, int out_size, void* d_ws, size_t ws_size,
                              hipStream_t stream) {
  const float* rays_o = (const float*)d_in[0];
  const float* rays_d = (const float*)d_in[1];
  const float* z_vals = (const float*)d_in[2];
  const float* sdf    = (const float*)d_in[3];
  const float* vmask  = (const float*)d_in[4];
  const float* inv_s  = (const float*)d_in[5];
  float* out = (float*)d_out;

  const int n_rays = in_sizes[2] / S;      // 131072
  const int blocks = n_rays / RPB;         // 8192
  hipLaunchKernelGGL(neus_upsample_kernel, dim3(blocks), dim3(TPB), 0, stream,
                     rays_o, rays_d, z_vals, sdf, vmask, inv_s, out);
}